// TokenChoiceTopKRouter_377957122267
// MI455X (gfx1250) — compile-verified
//
#include <hip/hip_runtime.h>
#include <math.h>

typedef __attribute__((ext_vector_type(2))) float v2f;
typedef __attribute__((ext_vector_type(4))) float v4f;
typedef __attribute__((ext_vector_type(8))) float v8f;
typedef __attribute__((ext_vector_type(4))) int   v4i;

// pointer-to-int4 in global (AS1) / LDS (AS3) address spaces
typedef __attribute__((address_space(1))) v4i* gv4i_p;
typedef __attribute__((address_space(3))) v4i* lv4i_p;

#define N_TOK  16384
#define DIM    2048
#define NEXP   64
#define TOPK   8
#define KCH    64            // K-chunk per pipeline stage
#define NCHUNK (DIM / KCH)   // 32
#define LPAD   68            // padded LDS row stride (floats); rows stay 16B-aligned

#define OFF_TOP   0
#define OFF_SCORE (N_TOK * TOPK)
#define OFF_IDX   (OFF_SCORE + N_TOK * NEXP)
#define OFF_HIST  (OFF_IDX + N_TOK * TOPK)
#define OFF_ENT   (OFF_HIST + NEXP)

#if __has_builtin(__builtin_amdgcn_global_load_async_to_lds_b128)
#define USE_ASYNC_LDS 1
#else
#define USE_ASYNC_LDS 0
#endif

#if __has_builtin(__builtin_amdgcn_s_wait_asynccnt)
#define WAIT_ASYNC() __builtin_amdgcn_s_wait_asynccnt(0)
#else
#define WAIT_ASYNC() asm volatile("s_wait_asynccnt 0" ::: "memory")
#endif

// Zero the histogram accumulator (atomicAdd target) every launch.
__global__ void router_init(float* __restrict__ out) {
    out[OFF_HIST + threadIdx.x] = 0.0f;
}

// One block = 16 tokens x 64 experts. 4 waves; wave w owns experts [16w,16w+16).
__global__ __launch_bounds__(128) void router_main(const float* __restrict__ x,
                                                   const float* __restrict__ gw,
                                                   const float* __restrict__ bias,
                                                   float* __restrict__ out,
                                                   float* __restrict__ ent_ws) {
    __shared__ float lx[2][16][LPAD];   // x tiles, double buffered (16 tok x 64 k)
    __shared__ float lw[2][64][LPAD];   // W tiles, double buffered (64 exp x 64 k)
    __shared__ float lsc[16][LPAD];     // sigmoid scores (16 tok x 64 exp)
    __shared__ float lbias[NEXP];

    const int tid   = threadIdx.x;
    const int lane  = tid & 31;
    const int wave  = tid >> 5;          // 0..3
    const int lrow  = lane & 15;         // row (A) / col (B) within 16x16 tile
    const int khalf = (lane >> 4) << 1;  // lanes 0-15 -> K{0,1}; 16-31 -> K{2,3}
    const int m0    = blockIdx.x * 16;

    if (tid < NEXP) lbias[tid] = bias[tid];

#if USE_ASYNC_LDS
    // Async DMA of one K-chunk into LDS buffer `buf`: 16B granules, coalesced.
    auto issue_chunk = [&](int buf, int k0) {
#pragma unroll
        for (int i = 0; i < 2; ++i) {        // x tile: 256 granules / 128 thr
            int q = i * 128 + tid;
            int r = q >> 4, c = (q & 15) << 2;
            __builtin_amdgcn_global_load_async_to_lds_b128(
                (gv4i_p)(x + (size_t)(m0 + r) * DIM + k0 + c),
                (lv4i_p)(&lx[buf][r][c]), 0, 0);
        }
#pragma unroll
        for (int i = 0; i < 8; ++i) {        // W tile: 1024 granules / 128 thr
            int q = i * 128 + tid;
            int r = q >> 4, c = (q & 15) << 2;
            __builtin_amdgcn_global_load_async_to_lds_b128(
                (gv4i_p)(gw + (size_t)r * DIM + k0 + c),
                (lv4i_p)(&lw[buf][r][c]), 0, 0);
        }
    };
#else
    // Fallback: synchronous 16B-wide staging through VGPRs.
    auto issue_chunk = [&](int buf, int k0) {
#pragma unroll
        for (int i = 0; i < 2; ++i) {
            int q = i * 128 + tid;
            int r = q >> 4, c = (q & 15) << 2;
            *(v4f*)&lx[buf][r][c] = *(const v4f*)(x + (size_t)(m0 + r) * DIM + k0 + c);
        }
#pragma unroll
        for (int i = 0; i < 8; ++i) {
            int q = i * 128 + tid;
            int r = q >> 4, c = (q & 15) << 2;
            *(v4f*)&lw[buf][r][c] = *(const v4f*)(gw + (size_t)r * DIM + k0 + c);
        }
    };
#endif

    v8f acc = {};

#if USE_ASYNC_LDS
    issue_chunk(0, 0);
    for (int c = 0; c < NCHUNK; ++c) {
        const int p = c & 1;
        WAIT_ASYNC();            // this wave's fills of buf[p] (and older) done
        __syncthreads();         // every wave's fills visible
        if (c + 1 < NCHUNK) issue_chunk(p ^ 1, (c + 1) * KCH);  // DMA overlaps WMMA
#pragma unroll
        for (int kk = 0; kk < KCH; kk += 4) {
            v2f a, b;
            a.x = lx[p][lrow][kk + khalf];
            a.y = lx[p][lrow][kk + khalf + 1];
            b.x = lw[p][wave * 16 + lrow][kk + khalf];
            b.y = lw[p][wave * 16 + lrow][kk + khalf + 1];
            acc = __builtin_amdgcn_wmma_f32_16x16x4_f32(
                false, a, false, b, (short)0, acc, false, false);
        }
        __syncthreads();         // all reads of buf[p] done before it is refilled
    }
#else
    for (int c = 0; c < NCHUNK; ++c) {
        const int p = c & 1;
        issue_chunk(p, c * KCH);
        __syncthreads();
#pragma unroll
        for (int kk = 0; kk < KCH; kk += 4) {
            v2f a, b;
            a.x = lx[p][lrow][kk + khalf];
            a.y = lx[p][lrow][kk + khalf + 1];
            b.x = lw[p][wave * 16 + lrow][kk + khalf];
            b.y = lw[p][wave * 16 + lrow][kk + khalf + 1];
            acc = __builtin_amdgcn_wmma_f32_16x16x4_f32(
                false, a, false, b, (short)0, acc, false, false);
        }
        __syncthreads();
    }
#endif

    // Sigmoid epilogue; C layout: VGPR r -> row r (lanes 0-15) / row r+8 (16-31).
    const int col = wave * 16 + lrow;
    for (int r = 0; r < 8; ++r) {
        int rl = (lane < 16) ? r : (r + 8);
        float s = 1.0f / (1.0f + __expf(-acc[r]));
        out[OFF_SCORE + (size_t)(m0 + rl) * NEXP + col] = s;
        lsc[rl][col] = s;
    }
    __syncthreads();

    // Biased top-8 per token; strict '>' argmax -> lowest index wins ties
    // (matches jax.lax.top_k).
    if (tid < 16) {
        const int t = m0 + tid;
        unsigned long long taken = 0ull;
        float tops[TOPK];
        int   idxs[TOPK];
        float ssum = 0.0f;
        for (int i = 0; i < TOPK; ++i) {
            float best = -INFINITY;
            int   bi   = 0;
            for (int e = 0; e < NEXP; ++e) {
                if (taken & (1ull << e)) continue;
                float v = lsc[tid][e] + lbias[e];
                if (v > best) { best = v; bi = e; }
            }
            taken  |= (1ull << bi);
            idxs[i] = bi;
            tops[i] = lsc[tid][bi];   // gather UNbiased score
            ssum   += tops[i];
        }
        const float inv = 1.0f / (ssum + 1e-20f);  // ROUTE_SCALE == 1
        float ent = 0.0f;
        for (int i = 0; i < TOPK; ++i) {
            float p = tops[i] * inv;
            out[OFF_TOP + (size_t)t * TOPK + i] = p;
            out[OFF_IDX + (size_t)t * TOPK + i] = (float)idxs[i];
            atomicAdd(&out[OFF_HIST + idxs[i]], 1.0f);  // exact +1.0 adds: deterministic
            ent += p * __logf(p);
        }
        ent_ws[t] = ent;   // per-token entropy term (deterministic reduce later)
    }
}

// Fixed-order tree reduction over 16384 per-token terms -> -mean.
__global__ __launch_bounds__(256) void router_reduce(const float* __restrict__ ent_ws,
                                                     float* __restrict__ out) {
    __shared__ float sh[256];
    const int tid = threadIdx.x;
    float s = 0.0f;
    for (int i = 0; i < N_TOK / 256; ++i) s += ent_ws[tid + i * 256];
    sh[tid] = s;
    __syncthreads();
    for (int off = 128; off > 0; off >>= 1) {
        if (tid < off) sh[tid] += sh[tid + off];
        __syncthreads();
    }
    if (tid == 0) out[OFF_ENT] = -sh[0] / (float)N_TOK;
}

extern "C" void kernel_launch(void* const* d_in, const int* in_sizes, int n_in,
                              void* d_out, int out_size, void* d_ws, size_t ws_size,
                              hipStream_t stream) {
    const float* x    = (const float*)d_in[0];  // [16384, 2048]
    const float* gw   = (const float*)d_in[1];  // [64, 2048]
    const float* bias = (const float*)d_in[2];  // [64]
    float* out    = (float*)d_out;
    float* ent_ws = (float*)d_ws;               // 16384 floats of scratch

    router_init<<<1, NEXP, 0, stream>>>(out);
    router_main<<<N_TOK / 16, 128, 0, stream>>>(x, gw, bias, out, ent_ws);
    router_reduce<<<1, 256, 0, stream>>>(ent_ws, out);
}